// Attention_14955076125471
// MI455X (gfx1250) — compile-verified
//
#include <hip/hip_runtime.h>
#include <math.h>

#define Hdim 128
#define SEQ 200000
#define TILE_ROWS 64
#define ROW_PITCH 132          // floats; 528B row stride: 16B aligned, bank-conflict free
#define NTILES 3125            // 200000 / 64 exactly -> NO tail tile
#define NB_EXP 782             // ceil(200000 / 256)

typedef __attribute__((ext_vector_type(2))) float v2f;
typedef __attribute__((ext_vector_type(8))) float v8f;

// ---------------- Kernel 1: u[h] = sum_o v[o]*W[o,H+h]; c0 = sum_o v[o]*(h0.W[o,:H] + b[o])
__global__ __launch_bounds__(128)
void prep_kernel(const float* __restrict__ hidden, const float* __restrict__ W,
                 const float* __restrict__ b, const float* __restrict__ v,
                 float* __restrict__ uc) {
    const int t = threadIdx.x;           // 0..127
    // u[t]: coalesced across threads each iteration
    float acc = 0.f;
    for (int o = 0; o < Hdim; ++o)
        acc += v[o] * W[o * (2 * Hdim) + Hdim + t];
    uc[t] = acc;
    // hproj[t] = hidden . W[t, :H]
    float hp = 0.f;
    for (int h = 0; h < Hdim; ++h)
        hp += hidden[h] * W[t * (2 * Hdim) + h];
    __shared__ float red[128];
    red[t] = v[t] * (hp + b[t]);
    __syncthreads();
    for (int s = 64; s > 0; s >>= 1) {
        if (t < s) red[t] += red[t + s];
        __syncthreads();
    }
    if (t == 0) uc[Hdim] = red[0];       // c0
}

// ---------------- Kernel 2: scores[s] = c0 + enc[s].u  via WMMA f32 16x16x4
// Tile staging uses GLOBAL_LOAD_ASYNC_TO_LDS_B128 (memory->LDS DMA, ASYNCcnt).
__global__ __launch_bounds__(128)
void scores_kernel(const float* __restrict__ enc, const float* __restrict__ uc,
                   float* __restrict__ out, float* __restrict__ bmax) {
    __shared__ float lds[TILE_ROWS * ROW_PITCH + Hdim + 128];
    float* uLds   = &lds[TILE_ROWS * ROW_PITCH];
    float* redArr = uLds + Hdim;

    const int t = threadIdx.x;                    // 0..127 (4 waves)
    const long long rowBase = (long long)blockIdx.x * TILE_ROWS;

    uLds[t] = uc[t];                              // u into LDS (t < 128 == Hdim)
    const float c0 = uc[Hdim];

    // Stage 64x128 f32 tile straight into LDS: 16 async B128 DMAs per thread,
    // fully coalesced (consecutive threads -> consecutive 16B chunks).
#pragma unroll
    for (int i = 0; i < 16; ++i) {
        int f  = t + i * 128;                     // 0..2047 float4 slots
        int r  = f >> 5;                          // row 0..63
        int c4 = f & 31;                          // float4 column
        unsigned lds_byte = (unsigned)((r * ROW_PITCH + c4 * 4) * 4);
        const float* gptr = enc + (rowBase + r) * Hdim + c4 * 4;
        asm volatile("global_load_async_to_lds_b128 %0, %1, off"
                     :
                     : "v"(lds_byte), "v"(gptr)
                     : "memory");
    }
    asm volatile("s_wait_asynccnt 0x0" ::: "memory");
    __syncthreads();

    // Per-wave WMMA: wave w handles rows w*16 .. w*16+15
    const int wave = t >> 5;
    const int lane = t & 31;
    const int m  = lane & 15;
    const int hi = lane >> 4;                     // lanes 16-31 carry K+2,K+3
    const float* aRow = &lds[(wave * 16 + m) * ROW_PITCH];

    v8f c = {};
#pragma unroll
    for (int k = 0; k < Hdim; k += 4) {
        // A(16x4): lanes0-15 = {K0,K1}, lanes16-31 = {K2,K3} (ISA 7.12.2)
        v2f a = *(const v2f*)(aRow + k + 2 * hi);
        // B(4x16): broadcast u chunk across all N columns; lane-half selects K-half
        v2f bq = *(const v2f*)(uLds + k + 2 * hi);
        c = __builtin_amdgcn_wmma_f32_16x16x4_f32(
                false, a, false, bq, (short)0, c, false, false);
    }

    __syncthreads();                              // done reading tile; reuse lds[0..63]
    // All D columns equal => lane0 holds scores m=0..7, lane16 holds m=8..15
    if (m == 0) {
#pragma unroll
        for (int r = 0; r < 8; ++r)
            lds[wave * 16 + hi * 8 + r] = c[r] + c0;
    }
    __syncthreads();

    float smax = -INFINITY;
    if (t < TILE_ROWS) {
        float s = lds[t];
        out[rowBase + t] = s;
        smax = s;
    }
    redArr[t] = smax;
    __syncthreads();
    for (int s = 64; s > 0; s >>= 1) {
        if (t < s) redArr[t] = fmaxf(redArr[t], redArr[t + s]);
        __syncthreads();
    }
    if (t == 0) bmax[blockIdx.x] = redArr[0];
}

// ---------------- Kernel 3: global max over block maxes (fixed-order, deterministic)
__global__ __launch_bounds__(256)
void reduce_max_kernel(const float* __restrict__ bmax, int n, float* __restrict__ gmax) {
    __shared__ float red[256];
    float m = -INFINITY;
    for (int i = threadIdx.x; i < n; i += 256) m = fmaxf(m, bmax[i]);
    red[threadIdx.x] = m;
    __syncthreads();
    for (int s = 128; s > 0; s >>= 1) {
        if (threadIdx.x < s) red[threadIdx.x] = fmaxf(red[threadIdx.x], red[threadIdx.x + s]);
        __syncthreads();
    }
    if (threadIdx.x == 0) *gmax = red[0];
}

// ---------------- Kernel 4: e = exp(score - M) in-place; per-block partial sums
__global__ __launch_bounds__(256)
void exp_kernel(float* __restrict__ out, const float* __restrict__ gmax,
                float* __restrict__ bsum) {
    __shared__ float red[256];
    const float M = *gmax;
    int idx = blockIdx.x * 256 + threadIdx.x;
    float e = 0.f;
    if (idx < SEQ) {
        e = expf(out[idx] - M);
        out[idx] = e;
    }
    red[threadIdx.x] = e;
    __syncthreads();
    for (int s = 128; s > 0; s >>= 1) {
        if (threadIdx.x < s) red[threadIdx.x] += red[threadIdx.x + s];
        __syncthreads();
    }
    if (threadIdx.x == 0) bsum[blockIdx.x] = red[0];
}

// ---------------- Kernel 5: total sum -> 1/S (fixed-order, deterministic)
__global__ __launch_bounds__(256)
void reduce_sum_kernel(const float* __restrict__ bsum, int n, float* __restrict__ invS) {
    __shared__ float red[256];
    float a = 0.f;
    for (int i = threadIdx.x; i < n; i += 256) a += bsum[i];
    red[threadIdx.x] = a;
    __syncthreads();
    for (int s = 128; s > 0; s >>= 1) {
        if (threadIdx.x < s) red[threadIdx.x] += red[threadIdx.x + s];
        __syncthreads();
    }
    if (threadIdx.x == 0) *invS = 1.0f / red[0];
}

// ---------------- Kernel 6: normalize
__global__ __launch_bounds__(256)
void scale_kernel(float* __restrict__ out, const float* __restrict__ invS) {
    int idx = blockIdx.x * 256 + threadIdx.x;
    if (idx < SEQ) out[idx] *= *invS;
}

extern "C" void kernel_launch(void* const* d_in, const int* in_sizes, int n_in,
                              void* d_out, int out_size, void* d_ws, size_t ws_size,
                              hipStream_t stream) {
    const float* hidden = (const float*)d_in[0];   // (1,1,128)
    const float* enc    = (const float*)d_in[1];   // (200000,1,128)
    const float* W      = (const float*)d_in[2];   // (128,256)
    const float* b      = (const float*)d_in[3];   // (128,)
    const float* v      = (const float*)d_in[4];   // (128,)
    float* out = (float*)d_out;                    // (200000,)
    float* ws  = (float*)d_ws;

    float* uc   = ws;          // [0..127]=u, [128]=c0
    float* gmax = ws + 129;
    float* invS = ws + 130;
    float* bmax = ws + 192;    // 3125 floats
    float* bsum = ws + 4096;   // 782 floats

    prep_kernel<<<1, 128, 0, stream>>>(hidden, W, b, v, uc);
    scores_kernel<<<NTILES, 128, 0, stream>>>(enc, uc, out, bmax);
    reduce_max_kernel<<<1, 256, 0, stream>>>(bmax, NTILES, gmax);
    exp_kernel<<<NB_EXP, 256, 0, stream>>>(out, gmax, bsum);
    reduce_sum_kernel<<<1, 256, 0, stream>>>(bsum, NB_EXP, invS);
    scale_kernel<<<NB_EXP, 256, 0, stream>>>(out, invS);
}